// FeatureOctree_38955353375209
// MI455X (gfx1250) — compile-verified
//
#include <hip/hip_runtime.h>

// FeatureOctree gather-interpolate for MI455X (gfx1250, wave32).
//
// Memory plan: the 96 MB feature table fits in the 192 MB L2, so feature
// gathers use default (RT) temporal hints to stay L2-resident, while the
// streaming inputs (indices, coords) and the output use non-temporal hints
// (TH_LOAD_NT / TH_STORE_NT) so they do not evict the table.
//
// Addressing plan: all per-lane addresses are formed as
//   uniform SGPR64 base + 32-bit unsigned VGPR byte offset (+ inst offset)
// so every access lowers to the GVS saddr form (global_load_b128 v, voff,
// s[base]) with one v_lshlrev_b32 per gather instead of 64-bit VALU adds.

typedef float __attribute__((ext_vector_type(4))) f32x4;
typedef int   __attribute__((ext_vector_type(4))) i32x4;

#define MAX_LEVEL   12
#define FEAT_LEVELS 3
#define FEAT_DIM    8
#define NPTS        1048576u
#define TABLE_N     1048577

__global__ __launch_bounds__(256) void octree_interp_kernel(
    const float* __restrict__ coord,     // [N, 3]
    const float* __restrict__ features,  // [3, TABLE_N, 8]
    const int*   __restrict__ indices,   // [3, N, 8]
    float*       __restrict__ out)       // [N, 8]
{
    const unsigned n      = blockIdx.x * blockDim.x + threadIdx.x;
    const unsigned nOff32 = n << 5;          // byte offset of this point's 32-B record

    // Streaming coordinate reads: non-temporal (do not pollute L2).
    const float* cp = coord + 3u * n;
    const float cx = __builtin_nontemporal_load(cp + 0);
    const float cy = __builtin_nontemporal_load(cp + 1);
    const float cz = __builtin_nontemporal_load(cp + 2);

    f32x4 accLo = {0.0f, 0.0f, 0.0f, 0.0f};
    f32x4 accHi = {0.0f, 0.0f, 0.0f, 0.0f};

#pragma unroll
    for (int i = 0; i < FEAT_LEVELS; ++i) {
        const float scale = (float)(1 << (MAX_LEVEL - i));   // 4096, 2048, 1024

        // coords = 2^level * (x*0.5 + 0.5); d = frac; smoothstep poly.
        float fx = __builtin_fmaf(cx, 0.5f, 0.5f) * scale;
        float fy = __builtin_fmaf(cy, 0.5f, 0.5f) * scale;
        float fz = __builtin_fmaf(cz, 0.5f, 0.5f) * scale;
        float dx = fx - __builtin_floorf(fx);
        float dy = fy - __builtin_floorf(fy);
        float dz = fz - __builtin_floorf(fz);
        const float tx = dx * dx * (3.0f - 2.0f * dx);       // 3d^2 - 2d^3
        const float ty = dy * dy * (3.0f - 2.0f * dy);
        const float tz = dz * dz * (3.0f - 2.0f * dz);
        const float ux = 1.0f - tx;
        const float uy = 1.0f - ty;
        const float uz = 1.0f - tz;

        // Corner c: bit2 -> x, bit1 -> y, bit0 -> z (reference stack order).
        float w[8];
        w[0] = ux * uy * uz;  w[1] = ux * uy * tz;
        w[2] = ux * ty * uz;  w[3] = ux * ty * tz;
        w[4] = tx * uy * uz;  w[5] = tx * uy * tz;
        w[6] = tx * ty * uz;  w[7] = tx * ty * tz;

        // Streaming index reads: saddr base (uniform per level) + n*32 offset.
        const char* ibase = (const char*)indices + (size_t)i * NPTS * 32u;
        const i32x4 idA = __builtin_nontemporal_load((const i32x4*)(ibase + nOff32));
        const i32x4 idB = __builtin_nontemporal_load((const i32x4*)(ibase + nOff32 + 16));
        const int id[8] = { idA.x, idA.y, idA.z, idA.w,
                            idB.x, idB.y, idB.z, idB.w };

        // L2-resident table gathers: uniform SGPR base + (id << 5) byte offset,
        // two b128 loads folding 0/16 into the instruction offset.
        const char* fbase =
            (const char*)features + (size_t)(FEAT_LEVELS - 1 - i) * (size_t)TABLE_N * 32u;

        f32x4 glo[8], ghi[8];
#pragma unroll
        for (int c = 0; c < 8; ++c) {
            const unsigned off = (unsigned)id[c] << 5;       // <= 2^25, fits u32
            const f32x4* g = (const f32x4*)(fbase + off);
            glo[c] = g[0];
            ghi[c] = g[1];
        }

        // Masked accumulate: row TABLE_N-1 is defined to be zero.
#pragma unroll
        for (int c = 0; c < 8; ++c) {
            const float wm = (id[c] == TABLE_N - 1) ? 0.0f : w[c];
            accLo += wm * glo[c];
            accHi += wm * ghi[c];
        }
    }

    // Streaming output: non-temporal b128 stores, saddr base + n*32 offset.
    char* obase = (char*)out;
    __builtin_nontemporal_store(accLo, (f32x4*)(obase + nOff32));
    __builtin_nontemporal_store(accHi, (f32x4*)(obase + nOff32 + 16));
}

extern "C" void kernel_launch(void* const* d_in, const int* in_sizes, int n_in,
                              void* d_out, int out_size, void* d_ws, size_t ws_size,
                              hipStream_t stream) {
    (void)in_sizes; (void)n_in; (void)d_ws; (void)ws_size; (void)out_size;
    const float* coord    = (const float*)d_in[0];
    const float* features = (const float*)d_in[1];
    const int*   indices  = (const int*)d_in[2];
    float*       out      = (float*)d_out;

    const int threads = 256;                       // 8 wave32 waves per block
    const int blocks  = (int)(NPTS / threads);     // 4096
    octree_interp_kernel<<<blocks, threads, 0, stream>>>(coord, features, indices, out);
}